// Spacial_IRNN_3616362463794
// MI455X (gfx1250) — compile-verified
//
#include <hip/hip_runtime.h>
#include <cstdint>
#include <cstddef>

// Problem constants (match the reference).
#define B_  8
#define C_  64
#define H_  256
#define W_  256
#define PSTR 257          // padded plane row stride (floats); 257 % 64 == 1 ->
                          // conflict-free for lane-per-row AND lane-per-column
#define CHW 16            // w-chunk width for horizontal output staging
#define LPAD (CHW + 1)    // 17-float staging row stride -> conflict-free

typedef unsigned int u32;
typedef u32 u32x4 __attribute__((ext_vector_type(4)));
typedef u32 u32x8 __attribute__((ext_vector_type(8)));

// ---- CDNA5 counter waits (builtin when available, asm fallback) ----------
#if defined(__has_builtin)
#  if __has_builtin(__builtin_amdgcn_s_wait_asynccnt)
#    define WAIT_ASYNC() __builtin_amdgcn_s_wait_asynccnt(0)
#  endif
#  if __has_builtin(__builtin_amdgcn_s_wait_tensorcnt)
#    define WAIT_TENSOR() __builtin_amdgcn_s_wait_tensorcnt(0)
#  endif
#endif
#ifndef WAIT_ASYNC
#  define WAIT_ASYNC() asm volatile("s_wait_asynccnt 0" ::: "memory")
#endif
#ifndef WAIT_TENSOR
#  define WAIT_TENSOR() asm volatile("s_wait_tensorcnt 0" ::: "memory")
#endif

// LDS byte offset of a shared-memory object: flat LDS aperture keeps the
// wave-relative LDS offset in addr[31:0] (ISA 10.2), so truncation works.
__device__ __forceinline__ unsigned lds_off(const void* p) {
  return (unsigned)(unsigned long long)p;
}

// =========================================================================
// Fused IRNN kernel: one block per (b,c) plane, 256 threads (8 wave32).
//  1) TDM tensor_load_to_lds DMAs the whole 256x256 fp32 plane into LDS,
//     hardware-padding +1 DWORD every 256 DWORDs -> 257-float row stride.
//  2) down/up scans: thread = column w, LDS reads conflict-free, global
//     stores coalesced 128B/wave.
//  3) right/left scans: thread = row h, LDS reads conflict-free; 256x16
//     result chunks staged in `tout` and written with
//     global_store_async_from_lds_b32 (coalesced by row).
// Input is read from HBM exactly once: 134MB in + 537MB out ~= 671MB total.
// =========================================================================
__global__ void __launch_bounds__(256)
irnn_fused(const float* __restrict__ in,
           const float* __restrict__ uw, const float* __restrict__ ub,
           const float* __restrict__ rw, const float* __restrict__ rb,
           const float* __restrict__ dw, const float* __restrict__ db,
           const float* __restrict__ lw, const float* __restrict__ lb,
           float* __restrict__ out) {
  __shared__ float plane[H_ * PSTR];   // 263,168 B
  __shared__ float tout [H_ * LPAD];   //  17,408 B   (total 280,576 B)

  const int bc  = blockIdx.x;          // b*C + c
  const int c   = bc & (C_ - 1);
  const int tid = threadIdx.x;

  const size_t planeElems = (size_t)H_ * W_;
  const float* gin = in + (size_t)bc * planeElems;

  // ---- 1) TDM plane load (wave 0 issues; TDM ignores EXEC) ----
  if (tid < 32) {
    const unsigned long long ga = (unsigned long long)(const void*)gin;
    u32x4 g0;
    g0.x = 1u;                                   // count=1, is_restore=0
    g0.y = lds_off(&plane[0]);                   // lds_addr
    g0.z = (u32)(ga & 0xFFFFFFFFu);              // global_addr[31:0]
    g0.w = (u32)((ga >> 32) & 0x1FFFFFFu)        // global_addr[56:32]
         | (2u << 30);                           // type = 2 ("image")
    u32x8 g1;
    g1.s0 = (2u << 16)                           // data_size = 4B
          | (1u << 20)                           // pad_enable
          | (7u << 22);                          // pad_interval: 256 DWORDs
                                                 // pad_amount = 0 -> 1 DWORD
    g1.s1 = (u32)W_ << 16;                       // tensor_dim0 = 256 (lo16)
    g1.s2 = (u32)H_ << 16;                       // dim0.hi=0 | tensor_dim1=256
    g1.s3 = (u32)W_ << 16;                       // dim1.hi=0 | tile_dim0=256
    g1.s4 = (u32)H_;                             // tile_dim1=256, tile_dim2=0
    g1.s5 = (u32)W_;                             // tensor_dim0_stride = 256
    g1.s6 = 0u;                                  // stride.hi | dim1_stride.lo
    g1.s7 = 0u;
    u32x4 gz = (u32x4)(0u);                      // groups 2/3: all-unused dims
    asm volatile("tensor_load_to_lds %0, %1, %2, %3"
                 :: "s"(g0), "s"(g1), "s"(gz), "s"(gz)
                 : "memory");
    WAIT_TENSOR();
  }
  __syncthreads();                               // publish plane to all waves

  // ---- 2) vertical scans (thread = column w) ----
  {
    const int w = tid;
    const float wd_ = dw[c], bd_ = db[c];
    const float wu_ = uw[c], bu_ = ub[c];
    float* gd = out + ((size_t)(2 * B_ * C_) + bc) * planeElems + w; // down
    float* gu = out + ((size_t)(0 * B_ * C_) + bc) * planeElems + w; // up

    float carry = 0.0f;
#pragma unroll 8
    for (int h = 0; h < H_; ++h) {
      carry = fmaxf(0.0f, fmaf(wd_, carry, plane[h * PSTR + w] + bd_));
      gd[(size_t)h * W_] = carry;
    }
    carry = 0.0f;
#pragma unroll 8
    for (int h = H_ - 1; h >= 0; --h) {
      carry = fmaxf(0.0f, fmaf(wu_, carry, plane[h * PSTR + w] + bu_));
      gu[(size_t)h * W_] = carry;
    }
  }

  // ---- 3) horizontal scans (thread = row h), staged output ----
  {
    const int h = tid;
    const float wr_ = rw[c], br_ = rb[c];
    const float wl_ = lw[c], bl_ = lb[c];
    float* gr = out + ((size_t)(1 * B_ * C_) + bc) * planeElems; // right
    float* gl = out + ((size_t)(3 * B_ * C_) + bc) * planeElems; // left
    const unsigned ltout = lds_off(&tout[0]);

    // right: w ascending
    float carry = 0.0f;
    for (int ck = 0; ck < W_ / CHW; ++ck) {
      const int w0 = ck * CHW;
#pragma unroll
      for (int wq = 0; wq < CHW; ++wq) {
        carry = fmaxf(0.0f, fmaf(wr_, carry, plane[h * PSTR + w0 + wq] + br_));
        tout[h * LPAD + wq] = carry;
      }
      __syncthreads();
#pragma unroll
      for (int j = 0; j < CHW; ++j) {
        const int q  = j * 256 + tid;
        const int hq = q >> 4, wq = q & 15;
        float* g = gr + hq * W_ + w0 + wq;
        const unsigned l = ltout + (unsigned)((hq * LPAD + wq) * 4);
        asm volatile("global_store_async_from_lds_b32 %0, %1, off"
                     :: "v"(g), "v"(l) : "memory");
      }
      WAIT_ASYNC();      // drain before tout is overwritten next chunk
      __syncthreads();
    }

    // left: w descending
    carry = 0.0f;
    for (int ck = W_ / CHW - 1; ck >= 0; --ck) {
      const int w0 = ck * CHW;
#pragma unroll
      for (int wq = CHW - 1; wq >= 0; --wq) {
        carry = fmaxf(0.0f, fmaf(wl_, carry, plane[h * PSTR + w0 + wq] + bl_));
        tout[h * LPAD + wq] = carry;
      }
      __syncthreads();
#pragma unroll
      for (int j = 0; j < CHW; ++j) {
        const int q  = j * 256 + tid;
        const int hq = q >> 4, wq = q & 15;
        float* g = gl + hq * W_ + w0 + wq;
        const unsigned l = ltout + (unsigned)((hq * LPAD + wq) * 4);
        asm volatile("global_store_async_from_lds_b32 %0, %1, off"
                     :: "v"(g), "v"(l) : "memory");
      }
      WAIT_ASYNC();
      __syncthreads();
    }
  }
}

// =========================================================================
// Launch
// =========================================================================
extern "C" void kernel_launch(void* const* d_in, const int* in_sizes, int n_in,
                              void* d_out, int out_size, void* d_ws, size_t ws_size,
                              hipStream_t stream) {
  // setup_inputs() order:
  // 0 input, 1 up_w, 2 right_w, 3 down_w, 4 left_w,
  // 5 up_b, 6 right_b, 7 down_b, 8 left_b
  const float* in      = (const float*)d_in[0];
  const float* up_w    = (const float*)d_in[1];
  const float* right_w = (const float*)d_in[2];
  const float* down_w  = (const float*)d_in[3];
  const float* left_w  = (const float*)d_in[4];
  const float* up_b    = (const float*)d_in[5];
  const float* right_b = (const float*)d_in[6];
  const float* down_b  = (const float*)d_in[7];
  const float* left_b  = (const float*)d_in[8];
  float* out = (float*)d_out;

  dim3 grid(B_ * C_);   // 512 blocks, one (b,c) plane each
  dim3 block(256);      // 8 wave32

  irnn_fused<<<grid, block, 0, stream>>>(in,
                                         up_w, up_b,
                                         right_w, right_b,
                                         down_w, down_b,
                                         left_w, left_b,
                                         out);
}